// ShiftingLayer_53798760349850
// MI455X (gfx1250) — compile-verified
//
#include <hip/hip_runtime.h>

// ShiftingLayer on MI455X (gfx1250, wave32):
//   out[b,d,c] = sum_s exp(-(s - center[b,d])^2) / AMP * input[b,s,c]
//   center[b,d] = (d + 1 + w[b,d]) * (S/D)
// The Gaussian underflows to exactly 0.0f in fp32 beyond |diff| ~ 10.2, so the
// contraction is exactly banded: walk only [min(center)-12, max(center)+12]
// per 16-row d-tile and contract with V_WMMA_F32_16X16X4_F32.

typedef __attribute__((ext_vector_type(2))) float v2f;
typedef __attribute__((ext_vector_type(8))) float v8f;

#define S_LEN 4096
#define C_LEN 256
#define D_LEN 1024
#define BAND  12.0f   // exp(-144) == 0.0f exactly; band is lossless in fp32

__global__ __launch_bounds__(256) void shifting_layer_wmma_kernel(
    const float* __restrict__ input,    // (16, 4096, 256) f32
    const float* __restrict__ weights,  // (16, 1024)      f32
    float* __restrict__ out)            // (16, 1024, 256) f32
{
    const int lane = threadIdx.x & 31;
    const int wave = threadIdx.x >> 5;      // 0..7, each owns 32 channels
    const int b    = blockIdx.x >> 6;       // 16 batches
    const int dt   = blockIdx.x & 63;       // 64 d-tiles of 16
    const int d0   = dt << 4;
    const int c0   = wave << 5;             // 2 WMMA c-tiles per wave

    const int m   = lane & 15;              // A row (both lane halves)
    const int n   = lane & 15;              // B/D column
    const int klo = (lane < 16) ? 0 : 2;    // K pair held by this lane half

    // Gaussian center for this lane's d-row.
    const float w      = weights[(size_t)b * D_LEN + d0 + m];
    const float center = ((float)(d0 + m + 1) + w) * ((float)S_LEN / (float)D_LEN);

    // Band union over the tile's 16 rows (wave32 butterfly; halves duplicate).
    float cmin = center, cmax = center;
    #pragma unroll
    for (int off = 16; off >= 1; off >>= 1) {
        cmin = fminf(cmin, __shfl_xor(cmin, off, 32));
        cmax = fmaxf(cmax, __shfl_xor(cmax, off, 32));
    }
    int s_start = (int)ceilf(cmin - BAND);
    int s_end   = (int)floorf(cmax + BAND);
    if (s_start < 1)     s_start = 1;
    if (s_end   > S_LEN) s_end   = S_LEN;

    const float invAmp = 1.0f / 1.772637204826652f;
    const float* inB = input + (size_t)b * S_LEN * C_LEN + (size_t)(c0 + n);

    v8f acc0 = {}, acc1 = {};

    if (s_start <= s_end) {                 // wave-uniform
        const float* p = inB + (size_t)(s_start - 1 + klo) * C_LEN;
        __builtin_prefetch(p, 0, 2);        // gfx1250 global_prefetch_b8

        float fk = (float)(s_start + klo) - center;   // diff at j=0
        int s0 = s_start;

        // Main loop: full chunks of 4, no guards, straight-line body.
        for (; s0 + 3 <= s_end; s0 += 4) {
            v2f amat, b0, b1;
            const float dj0 = fk;
            const float dj1 = fk + 1.0f;
            amat[0] = __expf(-dj0 * dj0) * invAmp;    // v_exp_f32 (TRANS)
            amat[1] = __expf(-dj1 * dj1) * invAmp;
            b0[0] = p[0];                              // 4 independent b32 loads
            b0[1] = p[C_LEN];
            b1[0] = p[16];
            b1[1] = p[C_LEN + 16];
            acc0 = __builtin_amdgcn_wmma_f32_16x16x4_f32(
                false, amat, false, b0, (short)0, acc0, false, false);
            acc1 = __builtin_amdgcn_wmma_f32_16x16x4_f32(
                false, amat, false, b1, (short)0, acc1, false, false);
            p  += 4 * C_LEN;
            fk += 4.0f;
        }

        // Tail chunk (<=3 rows; uniform branch). Branchless per-lane clamp:
        // load row clamped into range, zero the weight via v_cndmask --
        // EXEC stays all-ones for the WMMAs.
        if (s0 <= s_end) {
            v2f amat, b0, b1;
            #pragma unroll
            for (int j = 0; j < 2; ++j) {
                const int s_idx = s0 + klo + j;
                const int sr    = (s_idx <= s_end ? s_idx : s_end) - 1;
                const float diff = (float)s_idx - center;
                float aj = __expf(-diff * diff) * invAmp;
                if (s_idx > s_end) aj = 0.0f;          // select, not branch
                amat[j] = aj;
                b0[j] = inB[(size_t)sr * C_LEN];
                b1[j] = inB[(size_t)sr * C_LEN + 16];
            }
            acc0 = __builtin_amdgcn_wmma_f32_16x16x4_f32(
                false, amat, false, b0, (short)0, acc0, false, false);
            acc1 = __builtin_amdgcn_wmma_f32_16x16x4_f32(
                false, amat, false, b1, (short)0, acc1, false, false);
        }
    }

    // Store D tiles: VGPR r -> row (r or r+8), column n / n+16.
    float* outp = out + ((size_t)b * D_LEN + d0) * C_LEN + (size_t)(c0 + n);
    const int mbase = (lane < 16) ? 0 : 8;
    #pragma unroll
    for (int r = 0; r < 8; ++r) {
        outp[(size_t)(mbase + r) * C_LEN]      = acc0[r];
        outp[(size_t)(mbase + r) * C_LEN + 16] = acc1[r];
    }
}

extern "C" void kernel_launch(void* const* d_in, const int* in_sizes, int n_in,
                              void* d_out, int out_size, void* d_ws, size_t ws_size,
                              hipStream_t stream) {
    (void)in_sizes; (void)n_in; (void)d_ws; (void)ws_size; (void)out_size;
    const float* input   = (const float*)d_in[0];   // (16, 4096, 256)
    const float* weights = (const float*)d_in[1];   // (16, 1024)
    float*       out     = (float*)d_out;           // (16, 1024, 256)

    // 16 batches * 64 d-tiles = 1024 workgroups; 8 waves (256 thr) per WG,
    // each wave owns two 16-wide channel tiles sharing one A matrix.
    dim3 grid(16 * 64);
    dim3 block(256);
    shifting_layer_wmma_kernel<<<grid, block, 0, stream>>>(input, weights, out);
}